// CannyFilter_1202590843527
// MI455X (gfx1250) — compile-verified
//
#include <hip/hip_runtime.h>
#include <stdint.h>

// ---------------------------------------------------------------------------
// Canny pipeline for MI455X (gfx1250). Bandwidth-bound stencil: async
// global->LDS staging (CDNA5 ASYNCcnt path) + fused, register-blocked tiles.
// ---------------------------------------------------------------------------

#define USE_ASYNC_LDS 1

static constexpr int BATCH = 32, CH = 3, H = 512, W = 512;
static constexpr int TILE = 32;
static constexpr int HW = H * W;                        // 262144
static constexpr size_t NCHW = (size_t)BATCH * CH * HW; // 25165824
static constexpr size_t NHW = (size_t)BATCH * HW;       // 8388608

// ---- CDNA5 async global->LDS copy (4B per lane), tracked by ASYNCcnt ------
__device__ __forceinline__ void lds_copy(float* lds_ptr, const float* gptr) {
#if USE_ASYNC_LDS
  // Low 32 bits of a generic shared pointer == LDS byte offset on gfx1250.
  asm volatile("global_load_async_to_lds_b32 %0, %1, off"
               :
               : "v"((unsigned)(uintptr_t)lds_ptr),
                 "v"((unsigned long long)(uintptr_t)gptr)
               : "memory");
#else
  *lds_ptr = *gptr;
#endif
}

__device__ __forceinline__ void lds_copy_wait() {
#if USE_ASYNC_LDS
  asm volatile("s_wait_asynccnt 0" ::: "memory");
#endif
}

// ---------------------------------------------------------------------------
// Kernel 1: fused gaussian blur (depthwise) + sobel grad + magnitude/orient.
// One 32x32 tile per 256-thread block; img staged with halo=2, blur halo=1.
// Sobel sees ZERO outside the image (reference zero-pads the blurred map).
// ---------------------------------------------------------------------------
__global__ void __launch_bounds__(256) k_blur_grad(
    const float* __restrict__ img, const float* __restrict__ gauss,
    const float* __restrict__ sobel,
    float* __restrict__ o_blur, float* __restrict__ o_gx,
    float* __restrict__ o_gy, float* __restrict__ o_gm,
    float* __restrict__ o_or) {
  __shared__ float s_img[CH][36][36];
  __shared__ float s_blur[CH][34][34];

  const int tid = threadIdx.x;
  const int tx0 = blockIdx.x * TILE, ty0 = blockIdx.y * TILE;
  const int b = blockIdx.z;

  // 3x3 kernels: uniform address -> scalar loads, broadcast to all lanes.
  float gk[9], sk[9];
#pragma unroll
  for (int i = 0; i < 9; ++i) {
    gk[i] = gauss[i];
    sk[i] = sobel[i] * (1.0f / (float)CH);  // reference folds 1/C into sobel
  }

  // ---- Stage A: async-load img tile (+halo 2) with zero fill OOB ----
  const float* imgb = img + (size_t)b * CH * HW;
  for (int i = tid; i < CH * 36 * 36; i += 256) {
    const int c = i / (36 * 36), rem = i % (36 * 36);
    const int r = rem / 36, x = rem % 36;
    const int gy = ty0 - 2 + r, gx = tx0 - 2 + x;
    float* dst = &s_img[c][r][x];
    if ((unsigned)gy < (unsigned)H && (unsigned)gx < (unsigned)W)
      lds_copy(dst, imgb + (size_t)c * HW + (size_t)gy * W + gx);
    else
      *dst = 0.0f;
  }
  lds_copy_wait();
  __syncthreads();

  // ---- Stage B: blur into LDS (tile + halo 1); zero where outside image ----
  for (int i = tid; i < CH * 34 * 34; i += 256) {
    const int c = i / (34 * 34), rem = i % (34 * 34);
    const int r = rem / 34, x = rem % 34;
    const int gy = ty0 - 1 + r, gx = tx0 - 1 + x;
    float v = 0.0f;
    if ((unsigned)gy < (unsigned)H && (unsigned)gx < (unsigned)W) {
#pragma unroll
      for (int kr = 0; kr < 3; ++kr)
#pragma unroll
        for (int kc = 0; kc < 3; ++kc)
          v = fmaf(gk[kr * 3 + kc], s_img[c][r + kr][x + kc], v);
    }
    s_blur[c][r][x] = v;
  }
  __syncthreads();

  // ---- Stage B2: write blurred interior as aligned float4 rows ----
  for (int q = tid; q < CH * 32 * 8; q += 256) {
    const int c = q >> 8;           // / (32*8)
    const int rr = (q >> 3) & 31;
    const int xq = (q & 7) * 4;
    float4 v;
    v.x = s_blur[c][rr + 1][xq + 1];
    v.y = s_blur[c][rr + 1][xq + 2];
    v.z = s_blur[c][rr + 1][xq + 3];
    v.w = s_blur[c][rr + 1][xq + 4];
    *(float4*)&o_blur[(size_t)(b * CH + c) * HW + (size_t)(ty0 + rr) * W +
                      (tx0 + xq)] = v;
  }

  // ---- Stage C: 4 pixels per thread, 3x6 register window per (c,row) ----
  {
    const int r = tid >> 3;          // 0..31
    const int xq = (tid & 7) * 4;    // 0,4,...,28
    float gxv[4] = {0.f, 0.f, 0.f, 0.f};
    float gyv[4] = {0.f, 0.f, 0.f, 0.f};
#pragma unroll
    for (int c = 0; c < CH; ++c)
#pragma unroll
      for (int kr = 0; kr < 3; ++kr) {
        float w[6];
#pragma unroll
        for (int d = 0; d < 6; ++d) w[d] = s_blur[c][r + kr][xq + d];
#pragma unroll
        for (int j = 0; j < 4; ++j)
#pragma unroll
          for (int kc = 0; kc < 3; ++kc) {
            gxv[j] = fmaf(sk[kr * 3 + kc], w[j + kc], gxv[j]);  // sobel_x
            gyv[j] = fmaf(sk[kc * 3 + kr], w[j + kc], gyv[j]);  // sobel_x^T
          }
      }
    float4 vgx, vgy, vgm, vor;
    float* ogx = &vgx.x; float* ogy = &vgy.x;
    float* ogm = &vgm.x; float* oor = &vor.x;
#pragma unroll
    for (int j = 0; j < 4; ++j) {
      const float gm = sqrtf(gxv[j] * gxv[j] + gyv[j] * gyv[j]);
      // atan(gy/gx)*(360/pi)+180, rounded to multiple of 45 (RNE like jnp).
      // True division by 45: keeps multiples of 45 exact for later == tests.
      const float ang = atanf(gyv[j] / gxv[j]) * 114.59155902616465f + 180.0f;
      const float ori = rintf(ang / 45.0f) * 45.0f;
      ogx[j] = gxv[j];
      ogy[j] = gyv[j];
      ogm[j] = gm;
      oor[j] = ori;
    }
    const size_t o = (size_t)b * HW + (size_t)(ty0 + r) * W + (tx0 + xq);
    *(float4*)&o_gx[o] = vgx;
    *(float4*)&o_gy[o] = vgy;
    *(float4*)&o_gm[o] = vgm;
    *(float4*)&o_or[o] = vor;
  }
}

// ---------------------------------------------------------------------------
// Kernel 2: directional NMS + per-batch max of thin (pre-normalization).
// thin lands in the thin_edges output slice (rewritten in place by kernel 3).
// ---------------------------------------------------------------------------
__global__ void __launch_bounds__(256) k_nms_max(
    const float* __restrict__ gmG, const float* __restrict__ orG,
    float* __restrict__ thinOut, unsigned* __restrict__ bmax) {
  __shared__ float s_gm[34][34];
  __shared__ unsigned s_max;

  const int tid = threadIdx.x;
  const int tx0 = blockIdx.x * TILE, ty0 = blockIdx.y * TILE;
  const int b = blockIdx.z;
  if (tid == 0) s_max = 0u;

  const float* gmb = gmG + (size_t)b * HW;
  for (int i = tid; i < 34 * 34; i += 256) {
    const int r = i / 34, x = i % 34;
    const int gy = ty0 - 1 + r, gx = tx0 - 1 + x;
    float* dst = &s_gm[r][x];
    if ((unsigned)gy < (unsigned)H && (unsigned)gx < (unsigned)W)
      lds_copy(dst, gmb + (size_t)gy * W + gx);
    else
      *dst = 0.0f;  // directional conv zero-pads gm
  }
  lds_copy_wait();
  __syncthreads();

  // CCW directions i=0..7: (dr,dc), y-down coords (matches dir_k).
  const int offr[8] = {0, -1, -1, -1, 0, 1, 1, 1};
  const int offc[8] = {1, 1, 0, -1, -1, -1, 0, 1};

  const int r = tid >> 3;        // 0..31
  const int xq = (tid & 7) * 4;  // 0..28

  // 3x6 register window covers the 3x3 neighborhoods of 4 adjacent pixels.
  float w[3][6];
#pragma unroll
  for (int dr = 0; dr < 3; ++dr)
#pragma unroll
    for (int dc = 0; dc < 6; ++dc) w[dr][dc] = s_gm[r + dr][xq + dc];

  const size_t o = (size_t)b * HW + (size_t)(ty0 + r) * W + (tx0 + xq);
  const float4 ov = *(const float4*)&orG[o];
  const float orin[4] = {ov.x, ov.y, ov.z, ov.w};

  float4 tv;
  float* tp = &tv.x;
  float lmax = 0.0f;
#pragma unroll
  for (int j = 0; j < 4; ++j) {
    const float m = w[1][j + 1];
    // True division: orientation is an exact multiple of 45 -> exact index.
    const float fidx = fmodf(orin[j] / 45.0f, 8.0f);  // NaN stays NaN
    bool remove = false;
#pragma unroll
    for (int k = 0; k < 4; ++k) {
      const float d0 = m - w[1 + offr[k]][j + 1 + offc[k]];
      const float d1 = m - w[1 + offr[k + 4]][j + 1 + offc[k + 4]];
      const bool is_max = fminf(d0, d1) > 0.0f;
      const bool oriented = (fidx == (float)k) || (fidx == (float)(k + 4));
      remove = remove || (oriented && !is_max);
    }
    const float thin = remove ? 0.0f : m;
    tp[j] = thin;
    lmax = fmaxf(lmax, thin);
  }
  *(float4*)&thinOut[o] = tv;

  // wave32 shuffle reduce -> one LDS atomic per wave -> one global atomic.
#pragma unroll
  for (int s = 16; s > 0; s >>= 1) lmax = fmaxf(lmax, __shfl_xor(lmax, s, 32));
  if ((tid & 31) == 0) atomicMax(&s_max, __float_as_uint(lmax));  // thin >= 0
  __syncthreads();
  if (tid == 0) atomicMax(&bmax[b], s_max);
}

// ---------------------------------------------------------------------------
// Kernel 0: reset per-batch maxima (graph-replay safe, every call).
// ---------------------------------------------------------------------------
__global__ void k_init(unsigned* __restrict__ bmax) {
  if (threadIdx.x < BATCH) bmax[threadIdx.x] = 0u;
}

// ---------------------------------------------------------------------------
// Kernel 3: thin_edges = (thin / max_b) > threshold, in place, float4.
// ---------------------------------------------------------------------------
__global__ void __launch_bounds__(256) k_thresh(
    float* __restrict__ thin_edges, const unsigned* __restrict__ bmax,
    const float* __restrict__ thrP) {
  const float thr = thrP[0];
  const size_t i0 = ((size_t)blockIdx.x * 256 + threadIdx.x) * 4;
  if (i0 + 3 >= NHW) return;
  const int b = (int)(i0 / HW);  // HW % 4 == 0 -> all 4 lanes same batch
  const float mx = __uint_as_float(bmax[b]);
  float4 v = *(const float4*)(thin_edges + i0);
  float4 r;
  r.x = (v.x / mx > thr) ? 1.0f : 0.0f;
  r.y = (v.y / mx > thr) ? 1.0f : 0.0f;
  r.z = (v.z / mx > thr) ? 1.0f : 0.0f;
  r.w = (v.w / mx > thr) ? 1.0f : 0.0f;
  *(float4*)(thin_edges + i0) = r;
}

// ---------------------------------------------------------------------------
extern "C" void kernel_launch(void* const* d_in, const int* in_sizes, int n_in,
                              void* d_out, int out_size, void* d_ws,
                              size_t ws_size, hipStream_t stream) {
  (void)in_sizes; (void)n_in; (void)out_size; (void)ws_size;
  const float* img = (const float*)d_in[0];
  const float* thr = (const float*)d_in[1];
  const float* gk  = (const float*)d_in[2];
  const float* sk  = (const float*)d_in[3];
  // d_in[4] (dir_k) is a fixed algorithmic constant; directions hardcoded.

  float* out = (float*)d_out;
  float* o_blur = out;                 // [B,C,H,W]
  float* o_gx   = out + NCHW;          // [B,1,H,W]
  float* o_gy   = o_gx + NHW;
  float* o_gm   = o_gy + NHW;
  float* o_or   = o_gm + NHW;
  float* o_te   = o_or + NHW;          // thin (k2) then thin_edges (k3)

  unsigned* bmax = (unsigned*)d_ws;    // 32 uints

  k_init<<<1, 32, 0, stream>>>(bmax);

  dim3 grid(W / TILE, H / TILE, BATCH);
  k_blur_grad<<<grid, 256, 0, stream>>>(img, gk, sk, o_blur, o_gx, o_gy, o_gm,
                                        o_or);
  k_nms_max<<<grid, 256, 0, stream>>>(o_gm, o_or, o_te, bmax);

  const int nblk = (int)(NHW / (256 * 4));
  k_thresh<<<nblk, 256, 0, stream>>>(o_te, bmax, thr);
}